// LSTMCoordinator_82265803587816
// MI455X (gfx1250) — compile-verified
//
#include <hip/hip_runtime.h>

typedef float v2f __attribute__((ext_vector_type(2)));
typedef float v8f __attribute__((ext_vector_type(8)));

static __device__ __forceinline__ float fast_tanh(float x) {
#if __has_builtin(__builtin_amdgcn_tanhf)
    return __builtin_amdgcn_tanhf(x);           // V_TANH_F32
#else
    return fmaf(2.0f, __builtin_amdgcn_rcpf(1.0f + __expf(-2.0f * x)), -1.0f);
#endif
}
static __device__ __forceinline__ float fast_sigmoid(float x) {
#if __has_builtin(__builtin_amdgcn_tanhf)
    return fmaf(0.5f, __builtin_amdgcn_tanhf(0.5f * x), 0.5f);
#else
    return __builtin_amdgcn_rcpf(1.0f + __expf(-x));
#endif
}

// Tile: M = 16 samples, N = 128 padded gate cols (gate class gc at [gc*32, gc*32+19]).
// K padded to 24: rows 0-1 = x, 2-21 = hx, row 22 = fused bias (A supplies 1.0), row 23 = 0.
// Hidden unit h of sample m: lane (h%16 | half), acc tiles {h/16, +2, +4, +6} -> lane-local LSTM.
__global__ __launch_bounds__(256) void lstm_coord_kernel(
    const float* __restrict__ x,    const float* __restrict__ hx,
    const float* __restrict__ cx,   const float* __restrict__ u,
    const float* __restrict__ W_ih, const float* __restrict__ W_hh,
    const float* __restrict__ b_ih, const float* __restrict__ b_hh,
    const float* __restrict__ W_dec,const float* __restrict__ b_dec,
    float* __restrict__ out, int B)
{
    constexpr int H = 20, IN = 2, NPAD = 128, KS = 24, KP = 26; // KP: LDS row pad
    __shared__ __align__(16) float sWT[NPAD][KP];   // transposed: sWT[col][k]
    __shared__ __align__(16) float sWdec[64];
    __shared__ __align__(16) float sH[8][16][22];   // stride 22: even + bank-safe
    __shared__ float sBdec[4];

    const int tid  = threadIdx.x;
    const int lane = tid & 31;
    const int wv   = tid >> 5;
    const int c    = lane & 15;
    const int hi   = lane >> 4;

    // ---- stage transposed, padded weights (+ fused-bias K-row) into LDS ----
    for (int idx = tid; idx < NPAD * KS; idx += blockDim.x) {
        int n = idx / KS, k = idx % KS;
        int gc = n >> 5, hh = n & 31;
        float v = 0.0f;
        if (hh < H) {
            int r = gc * H + hh;                  // gate row, PyTorch i,f,g,o order
            if (k < IN)            v = W_ih[r * IN + k];
            else if (k < IN + H)   v = W_hh[r * H + (k - IN)];
            else if (k == IN + H)  v = b_ih[r] + b_hh[r];   // bias row (Z row = 1.0)
        }
        sWT[n][k] = v;
    }
    if (tid < 60) sWdec[tid] = W_dec[tid];
    if (tid < 3)  sBdec[tid] = b_dec[tid];
    __syncthreads();

    const int NT         = B >> 4;
    const int totalWaves = gridDim.x * (blockDim.x >> 5);
    const int gwave      = blockIdx.x * (blockDim.x >> 5) + wv;

    for (int t = gwave; t < NT; t += totalWaves) {
        const int m0 = t << 4;
        const int sA = m0 + c;                    // this lane's A-fragment row
        {
            int tn = t + totalWaves;              // prefetch next tile's hx row
            if (tn < NT) __builtin_prefetch(hx + (size_t)((tn << 4) + c) * H, 0, 1);
        }
        const float* zh = hx + (size_t)sA * H;

        // A fragments: each is one aligned b64 load. a[k] = (Z[4k+2hi], Z[4k+2hi+1])
        v2f a[6];
        {
            const float* p0 = hi ? zh : (x + (size_t)sA * IN);   // Z0,Z1 | Z2,Z3
            a[0] = *(const v2f*)p0;
        }
        #pragma unroll
        for (int k = 1; k < 5; ++k)
            a[k] = *(const v2f*)(zh + (4 * k - 2 + 2 * hi));
        {
            v2f t18 = *(const v2f*)(zh + 18);     // Z20,Z21 for lo half
            a[5][0] = hi ? 1.0f : t18[0];         // hi half: bias row = 1, pad = 0
            a[5][1] = hi ? 0.0f : t18[1];
        }

        // ---- GEMM: 8 N-tiles x 6 K-chunks; C starts as inline 0 (bias folded) ----
        v8f acc[8];
        #pragma unroll
        for (int j = 0; j < 8; ++j) {
            v2f b = *(const v2f*)&sWT[j * 16 + c][2 * hi];       // k-chunk 0
            v8f z = {};
            acc[j] = __builtin_amdgcn_wmma_f32_16x16x4_f32(
                false, a[0], false, b, (short)0, z, false, false);
        }
        #pragma unroll
        for (int k = 1; k < 6; ++k) {
            #pragma unroll
            for (int j = 0; j < 8; ++j) {
                v2f b = *(const v2f*)&sWT[j * 16 + c][4 * k + 2 * hi];
                acc[j] = __builtin_amdgcn_wmma_f32_16x16x4_f32(
                    false, a[k], false, b, (short)0, acc[j], false, false);
            }
        }

        // ---- pointwise LSTM stage (lane-local) ----
        float* outH = out + (size_t)2 * B;        // out: [action | logp | h | c]
        float* outC = out + (size_t)22 * B;
        #pragma unroll
        for (int r = 0; r < 8; ++r) {
            int sm = r + (hi << 3);               // acc row M = sample within tile
            size_t s = (size_t)(m0 + sm);
            float i1 = fast_sigmoid(acc[0][r]);
            float f1 = fast_sigmoid(acc[2][r]);
            float g1 = fast_tanh   (acc[4][r]);
            float o1 = fast_sigmoid(acc[6][r]);
            float cn = fmaf(f1, cx[s * H + c], i1 * g1);
            float hn = o1 * fast_tanh(cn);
            outH[s * H + c] = hn;
            outC[s * H + c] = cn;
            sH[wv][sm][c] = hn;
            if (c < 4) {                          // hidden units 16..19
                int h2 = 16 + c;
                float i2 = fast_sigmoid(acc[1][r]);
                float f2 = fast_sigmoid(acc[3][r]);
                float g2 = fast_tanh   (acc[5][r]);
                float o2 = fast_sigmoid(acc[7][r]);
                float cn2 = fmaf(f2, cx[s * H + h2], i2 * g2);
                float hn2 = o2 * fast_tanh(cn2);
                outH[s * H + h2] = hn2;
                outC[s * H + h2] = cn2;
                sH[wv][sm][h2] = hn2;
            }
        }
        __syncthreads();

        // ---- decode + log-softmax + inverse-CDF categorical sample ----
        if (lane < 16) {
            size_t s = (size_t)(m0 + lane);
            float l0 = sBdec[0], l1 = sBdec[1], l2 = sBdec[2];
            #pragma unroll
            for (int kk = 0; kk < 10; ++kk) {
                v2f hv = *(const v2f*)&sH[wv][lane][2 * kk];
                v2f w0 = *(const v2f*)&sWdec[2 * kk];
                v2f w1 = *(const v2f*)&sWdec[20 + 2 * kk];
                v2f w2 = *(const v2f*)&sWdec[40 + 2 * kk];
                l0 = fmaf(hv[1], w0[1], fmaf(hv[0], w0[0], l0));
                l1 = fmaf(hv[1], w1[1], fmaf(hv[0], w1[0], l1));
                l2 = fmaf(hv[1], w2[1], fmaf(hv[0], w2[0], l2));
            }
            float mx  = fmaxf(l0, fmaxf(l1, l2));
            float e0  = __expf(l0 - mx), e1 = __expf(l1 - mx), e2 = __expf(l2 - mx);
            float Zs  = e0 + e1 + e2;
            float lse = mx + __logf(Zs);
            float lp0 = l0 - lse, lp1 = l1 - lse, lp2 = l2 - lse;
            float iZ  = __builtin_amdgcn_rcpf(Zs);
            float cdf0 = e0 * iZ;
            float cdf1 = fmaf(e1, iZ, cdf0);
            float cdf2 = fmaf(e2, iZ, cdf1);
            float uu = u[s];
            int act = (cdf0 < uu) + (cdf1 < uu) + (cdf2 < uu);
            act = act > 2 ? 2 : act;
            float sel = (act == 0) ? lp0 : ((act == 1) ? lp1 : lp2);
            ((int*)out)[s]     = act;             // int32 action bit-pattern
            out[(size_t)B + s] = sel;             // selected log-prob
        }
        __syncthreads();                          // protect sH before next iter
    }
}

extern "C" void kernel_launch(void* const* d_in, const int* in_sizes, int n_in,
                              void* d_out, int out_size, void* d_ws, size_t ws_size,
                              hipStream_t stream) {
    (void)n_in; (void)out_size; (void)d_ws; (void)ws_size;
    const float* x     = (const float*)d_in[0];
    const float* hx    = (const float*)d_in[1];
    const float* cx    = (const float*)d_in[2];
    const float* u     = (const float*)d_in[3];
    const float* W_ih  = (const float*)d_in[4];
    const float* W_hh  = (const float*)d_in[5];
    const float* b_ih  = (const float*)d_in[6];
    const float* b_hh  = (const float*)d_in[7];
    const float* W_dec = (const float*)d_in[8];
    const float* b_dec = (const float*)d_in[9];

    const int B  = in_sizes[3];                   // u is [B,1]
    const int NT = B >> 4;
    int grid = 1024;                              // enforce uniform trip counts
    while (grid > 1 && (NT % (grid * 8)) != 0) grid >>= 1;

    lstm_coord_kernel<<<grid, 256, 0, stream>>>(
        x, hx, cx, u, W_ih, W_hh, b_ih, b_hh, W_dec, b_dec, (float*)d_out, B);
}